// HeteroLayerG_23192823399227
// MI455X (gfx1250) — compile-verified
//
#include <hip/hip_runtime.h>
#include <hip/hip_bf16.h>

typedef __attribute__((ext_vector_type(2))) float v2f;
typedef __attribute__((ext_vector_type(8))) float v8f;

#define DH 128   // hidden dim (DIN == DOUT == 128)

// ---------------------------------------------------------------------------
// C[M x 128] = A[M x 128] @ W[128 x 128] + bias[128]
// One block: 16 rows x 128 cols. 8 waves; wave w computes the 16x16 tile at
// columns [16w, 16w+16). K-loop: 32 x V_WMMA_F32_16X16X4_F32.
// A-tile staged in LDS (stride 132 to avoid bank conflicts); W from global
// (64KB, lives in L2 / WGP$).
// ---------------------------------------------------------------------------
__global__ __launch_bounds__(256) void gemm_bias_wmma(
    const float* __restrict__ A, const float* __restrict__ W,
    const float* __restrict__ bias, float* __restrict__ C, int M) {
  __shared__ float sA[16 * 132];

  const int m0  = blockIdx.x * 16;
  const int tid = threadIdx.x;

  // Stage 16x128 A tile into LDS (rows past M load as zero).
  for (int i = tid; i < 16 * DH; i += 256) {
    const int r = i >> 7;       // row 0..15
    const int c = i & 127;      // col 0..127
    float v = 0.0f;
    if (m0 + r < M) v = A[(size_t)(m0 + r) * DH + c];
    sA[r * 132 + c] = v;
  }
  __syncthreads();

  const int wave  = tid >> 5;
  const int lane  = tid & 31;
  const int n0    = wave << 4;
  const int m     = lane & 15;          // A row within tile (both half-waves)
  const int n     = n0 + (lane & 15);   // C column
  const int khalf = (lane >> 4) << 1;   // 0 for lanes 0-15, 2 for lanes 16-31

  v8f c = {};
#pragma unroll
  for (int k0 = 0; k0 < DH; k0 += 4) {
    const int ka = k0 + khalf;
    // A 16x4 f32 layout: lane m: VGPR0=A[m][ka], VGPR1=A[m][ka+1]
    v2f a;
    a.x = sA[m * 132 + ka];
    a.y = sA[m * 132 + ka + 1];
    // B 4x16 f32 layout: lane n: VGPR0=W[ka][n], VGPR1=W[ka+1][n]
    v2f b;
    b.x = W[(size_t)ka * DH + n];
    b.y = W[(size_t)(ka + 1) * DH + n];
    c = __builtin_amdgcn_wmma_f32_16x16x4_f32(
        /*neg_a=*/false, a, /*neg_b=*/false, b,
        /*c_mod=*/(short)0, c, /*reuse_a=*/false, /*reuse_b=*/false);
  }

  // C/D layout: lanes 0-15 VGPR r -> M=r ; lanes 16-31 VGPR r -> M=8+r
  const float bv    = bias[n];
  const int   rbase = m0 + ((lane >> 4) << 3);
#pragma unroll
  for (int r = 0; r < 8; ++r) {
    const int row = rbase + r;
    if (row < M) C[(size_t)row * DH + n] = c[r] + bv;
  }
}

// ---------------------------------------------------------------------------
// Scatter-add: 32 lanes per edge, each lane handles 4 consecutive columns.
// sum[dst] += h[src] * w ; deg[dst] += 1
// ---------------------------------------------------------------------------
__global__ __launch_bounds__(256) void edge_scatter(
    const float* __restrict__ h, const int* __restrict__ src,
    const int* __restrict__ dst, const float* __restrict__ w,
    float* __restrict__ sum, float* __restrict__ deg, int E) {
  const long long idx = (long long)blockIdx.x * blockDim.x + threadIdx.x;
  const int e = (int)(idx >> 5);
  const int l = (int)(idx & 31);
  if (e >= E) return;
  const int   s  = src[e];
  const int   d  = dst[e];
  const float we = w[e];
  const float4 hv = ((const float4*)(h + (size_t)s * DH))[l];
  float* out = sum + (size_t)d * DH + l * 4;
  atomicAdd(out + 0, hv.x * we);
  atomicAdd(out + 1, hv.y * we);
  atomicAdd(out + 2, hv.z * we);
  atomicAdd(out + 3, hv.w * we);
  if (l == 0) atomicAdd(deg + d, 1.0f);
}

// out[i][:] = sum[i][:] / max(deg[i], 1)
__global__ __launch_bounds__(256) void seg_divide(
    const float* __restrict__ sum, const float* __restrict__ deg,
    float* __restrict__ out, int N) {
  const long long idx = (long long)blockIdx.x * blockDim.x + threadIdx.x;
  if (idx >= (long long)N * DH) return;
  const int i = (int)(idx >> 7);
  const float dv = fmaxf(deg[i], 1.0f);
  out[idx] = sum[idx] / dv;
}

// doc = sumA/max(degA,1) + sumB/max(degB,1)
__global__ __launch_bounds__(256) void seg_divide_add2(
    const float* __restrict__ sumA, const float* __restrict__ degA,
    const float* __restrict__ sumB, const float* __restrict__ degB,
    float* __restrict__ out, int N) {
  const long long idx = (long long)blockIdx.x * blockDim.x + threadIdx.x;
  if (idx >= (long long)N * DH) return;
  const int i = (int)(idx >> 7);
  out[idx] = sumA[idx] / fmaxf(degA[i], 1.0f) + sumB[idx] / fmaxf(degB[i], 1.0f);
}

__global__ __launch_bounds__(256) void zero_f32(float* __restrict__ p, long long n) {
  const long long idx = (long long)blockIdx.x * blockDim.x + threadIdx.x;
  if (idx < n) p[idx] = 0.0f;
}

// ---------------------------------------------------------------------------
static inline void launch_zero(float* p, long long n, hipStream_t s) {
  zero_f32<<<(unsigned)((n + 255) / 256), 256, 0, s>>>(p, n);
}
static inline void launch_gemm(const float* A, const float* W, const float* b,
                               float* C, int M, hipStream_t s) {
  gemm_bias_wmma<<<(M + 15) / 16, 256, 0, s>>>(A, W, b, C, M);
}
static inline void launch_scatter(const float* h, const int* src, const int* dst,
                                  const float* w, float* sum, float* deg, int E,
                                  hipStream_t s) {
  const long long threads = (long long)E * 32;
  edge_scatter<<<(unsigned)((threads + 255) / 256), 256, 0, s>>>(h, src, dst, w,
                                                                sum, deg, E);
}

extern "C" void kernel_launch(void* const* d_in, const int* in_sizes, int n_in,
                              void* d_out, int out_size, void* d_ws, size_t ws_size,
                              hipStream_t stream) {
  // Inputs (setup_inputs order)
  const float* h_word  = (const float*)d_in[0];
  const float* h_topic = (const float*)d_in[1];
  const int*   ww_src  = (const int*)d_in[2];
  const int*   ww_dst  = (const int*)d_in[3];
  const float* ww_w    = (const float*)d_in[4];
  // wt edges (d_in[5..7]) are unused by the reference
  const int*   wd_src  = (const int*)d_in[8];
  const int*   wd_dst  = (const int*)d_in[9];
  const float* wd_w    = (const float*)d_in[10];
  const int*   td_src  = (const int*)d_in[11];
  const int*   td_dst  = (const int*)d_in[12];
  const float* td_w    = (const float*)d_in[13];
  const int*   tt_src  = (const int*)d_in[14];
  const int*   tt_dst  = (const int*)d_in[15];
  const float* tt_w    = (const float*)d_in[16];
  const float* W_ww = (const float*)d_in[17];
  const float* b_ww = (const float*)d_in[18];
  const float* W_wt = (const float*)d_in[19];
  const float* b_wt = (const float*)d_in[20];
  const float* W_wd = (const float*)d_in[21];
  const float* b_wd = (const float*)d_in[22];
  const float* W_td = (const float*)d_in[23];
  const float* b_td = (const float*)d_in[24];
  const float* W_tt = (const float*)d_in[25];
  const float* b_tt = (const float*)d_in[26];

  const int NW = in_sizes[0] / DH;
  const int NT = in_sizes[1] / DH;
  const int E_ww = in_sizes[2];
  const int E_wd = in_sizes[8];
  const int E_td = in_sizes[11];
  const int E_tt = in_sizes[14];
  const int ND = out_size / DH - NW - NT;

  // Output layout: word_out | topic_out | doc_out
  float* word_out  = (float*)d_out;
  float* topic_out = word_out + (size_t)NW * DH;
  float* doc_out   = topic_out + (size_t)NT * DH;

  // Workspace layout (fp32)
  float* ws    = (float*)d_ws;
  float* buf0  = ws;                          ws += (size_t)NW * DH;  // wh / wt tmp
  float* buf1  = ws;                          ws += (size_t)NW * DH;  // ww sums -> h_word1
  float* degW  = ws;                          ws += (size_t)NW;
  float* bufT0 = ws;                          ws += (size_t)NT * DH;  // td intermediate
  float* bufT1 = ws;                          ws += (size_t)NT * DH;  // h_topic2
  float* sumTT = ws;                          ws += (size_t)NT * DH;
  float* degTT = ws;                          ws += (size_t)NT;
  float* sumWD = ws;                          ws += (size_t)ND * DH;
  float* degWD = ws;                          ws += (size_t)ND;
  float* sumTD = ws;                          ws += (size_t)ND * DH;
  float* degTD = ws;                          ws += (size_t)ND;
  (void)ws_size; (void)n_in;

  // --- Stage 1: wh = h_word @ W_ww + b_ww ; h_word1 = seg_mean(wh, ww) ---
  launch_zero(buf1, (long long)NW * DH, stream);
  launch_zero(degW, NW, stream);
  launch_gemm(h_word, W_ww, b_ww, buf0, NW, stream);
  launch_scatter(buf0, ww_src, ww_dst, ww_w, buf1, degW, E_ww, stream);
  {
    const long long n = (long long)NW * DH;
    seg_divide<<<(unsigned)((n + 255) / 256), 256, 0, stream>>>(buf1, degW, buf1, NW);
  }

  // --- Stage 2 dense: h_word2 = (h_word1 @ W_wt + b) @ W_wd + b -> word_out
  launch_gemm(buf1, W_wt, b_wt, buf0, NW, stream);
  launch_gemm(buf0, W_wd, b_wd, word_out, NW, stream);
  //          h_topic2 = (h_topic @ W_td + b) @ W_tt + b
  launch_gemm(h_topic, W_td, b_td, bufT0, NT, stream);
  launch_gemm(bufT0, W_tt, b_tt, bufT1, NT, stream);

  // --- Stage 2 sparse ---
  launch_zero(sumTT, (long long)NT * DH, stream);
  launch_zero(degTT, NT, stream);
  launch_zero(sumWD, (long long)ND * DH, stream);
  launch_zero(degWD, ND, stream);
  launch_zero(sumTD, (long long)ND * DH, stream);
  launch_zero(degTD, ND, stream);

  launch_scatter(bufT1, tt_src, tt_dst, tt_w, sumTT, degTT, E_tt, stream);
  launch_scatter(word_out, wd_src, wd_dst, wd_w, sumWD, degWD, E_wd, stream);
  launch_scatter(bufT1, td_src, td_dst, td_w, sumTD, degTD, E_td, stream);

  {
    const long long n = (long long)NT * DH;
    seg_divide<<<(unsigned)((n + 255) / 256), 256, 0, stream>>>(sumTT, degTT, topic_out, NT);
  }
  {
    const long long n = (long long)ND * DH;
    seg_divide_add2<<<(unsigned)((n + 255) / 256), 256, 0, stream>>>(
        sumWD, degWD, sumTD, degTD, doc_out, ND);
  }
}